// SConv_2997887172766
// MI455X (gfx1250) — compile-verified
//
#include <hip/hip_runtime.h>

typedef __attribute__((ext_vector_type(2))) float v2f;
typedef __attribute__((ext_vector_type(8))) float v8f;

// reflect index into [0,7) for pad=3 (jnp.pad mode="reflect")
__device__ __forceinline__ int refl7(int q) {
    q = q < 0 ? -q : q;
    return q > 6 ? 12 - q : q;
}

// GEMM dims: C[M=64, N=196] = A[64, K=25088] x B[25088, 196]
//   K padded per-ic: 49 -> 52 (zeros), N padded 196 -> 208.
#define KPAD   52
#define NPAD   208
#define NSPLIT 16              // split-K factor (32 ics per split)
#define ICS_PER_SPLIT (512 / NSPLIT)

// ---------------------------------------------------------------------------
// Prep A: Apad[ic][52][64] = w_key[m, ic, r] (r<49) else 0
// ---------------------------------------------------------------------------
__global__ __launch_bounds__(256)
void prep_a_kernel(const float* __restrict__ wk, float* __restrict__ Apad)
{
    const int idx = blockIdx.x * 256 + threadIdx.x;   // over 512*52*64
    if (idx >= 512 * KPAD * 64) return;
    const int m  = idx & 63;
    const int t  = idx >> 6;          // ic*52 + r
    const int r  = t % KPAD;
    const int ic = t / KPAD;
    Apad[idx] = (r < 49) ? wk[m * 25088 + ic * 49 + r] : 0.f;
}

// ---------------------------------------------------------------------------
// Prep B: Bpad[ic][52][208] = reflect-gathered im2col of representation
//   n = bb*49 + iy*7 + ix ;  r = ky*7 + kx
//   value = rep[bb, ic, refl(iy+ky-3), refl(ix+kx-3)] ; zero in all pad slots
// ---------------------------------------------------------------------------
__global__ __launch_bounds__(256)
void prep_b_kernel(const float* __restrict__ rep, float* __restrict__ Bpad)
{
    const int idx = blockIdx.x * 256 + threadIdx.x;   // over 512*52*208
    if (idx >= 512 * KPAD * NPAD) return;
    const int n  = idx % NPAD;
    const int t  = idx / NPAD;        // ic*52 + r
    const int r  = t % KPAD;
    const int ic = t / KPAD;
    float v = 0.f;
    if (r < 49 && n < 196) {
        const int bb = n / 49, rn = n % 49;
        const int iy = rn / 7, ix = rn % 7;
        const int ky = r / 7,  kx = r % 7;
        v = rep[(bb * 512 + ic) * 49 + refl7(iy + ky - 3) * 7 + refl7(ix + kx - 3)];
    }
    Bpad[idx] = v;
}

// ---------------------------------------------------------------------------
// Split-K WMMA GEMM: one wave per (split, 16x16 C tile).
// Inner step = 4 coalesced global_load_b32 (immediate offsets) + 1 WMMA.
// A fragment (16x4 f32, ISA layout): lane holds M = lane%16, K pair 2*(lane/16).
// C fragment: VGPR v -> M = v + 8*(lane/16), N = lane%16.
// ---------------------------------------------------------------------------
__global__ __launch_bounds__(32)
void keymap_wmma_kernel(const float* __restrict__ Apad,     // [512][52][64]
                        const float* __restrict__ Bpad,     // [512][52][208]
                        float* __restrict__ partial)        // [16][64][208]
{
    const int tile  = blockIdx.x % 52;
    const int split = blockIdx.x / 52;        // 0..15
    const int mt    = tile & 3;               // M tile (4)
    const int nt    = tile >> 2;              // N tile (13)
    const int lane  = threadIdx.x;
    const int half  = lane >> 4;              // K-pair selector
    const int lm    = lane & 15;
    const int mA    = mt * 16 + lm;           // A row for this lane
    const int nB    = nt * 16 + lm;           // B/C column for this lane

    const int ic0 = split * ICS_PER_SPLIT;
    // per-lane base pointers; all K-step offsets become immediates
    const float* aBase = Apad + (size_t)ic0 * KPAD * 64   + half * 2 * 64   + mA;
    const float* bBase = Bpad + (size_t)ic0 * KPAD * NPAD + half * 2 * NPAD + nB;

    v8f acc = {};
    for (int ic = 0; ic < ICS_PER_SPLIT; ++ic) {
        const float* ap = aBase + ic * (KPAD * 64);
        const float* bp = bBase + ic * (KPAD * NPAD);
#pragma unroll
        for (int s = 0; s < 13; ++s) {        // 52 padded K per ic, 4 per WMMA
            v2f A, B;
            A.x = ap[(s * 4) * 64];
            A.y = ap[(s * 4 + 1) * 64];
            B.x = bp[(s * 4) * NPAD];
            B.y = bp[(s * 4 + 1) * NPAD];
            // 8 args: (neg_a, A, neg_b, B, c_mod, C, reuse_a, reuse_b)
            acc = __builtin_amdgcn_wmma_f32_16x16x4_f32(
                false, A, false, B, (short)0, acc, false, false);
        }
    }

#pragma unroll
    for (int v = 0; v < 8; ++v) {
        const int m = mt * 16 + v + 8 * half;
        partial[((size_t)split * 64 + m) * NPAD + nB] = acc[v];
    }
}

// ---------------------------------------------------------------------------
// Reduce split-K partials + bias + LeakyReLU(0.1) -> key_map[4,64,7,7]
// ---------------------------------------------------------------------------
__global__ __launch_bounds__(256)
void reduce_kernel(const float* __restrict__ partial,   // [16][64][208]
                   const float* __restrict__ bk,        // [64]
                   float* __restrict__ km)              // [4,64,7,7]
{
    const int idx = blockIdx.x * 256 + threadIdx.x;     // over 64*208
    if (idx >= 64 * NPAD) return;
    const int n = idx % NPAD;
    const int m = idx / NPAD;
    if (n >= 196) return;
    float s = 0.f;
#pragma unroll
    for (int sp = 0; sp < NSPLIT; ++sp)
        s += partial[((size_t)sp * 64 + m) * NPAD + n];
    s += bk[m];
    s = s >= 0.f ? s : 0.1f * s;
    const int bb = n / 49, rn = n % 49;
    km[(bb * 64 + m) * 49 + rn] = s;
}

// ---------------------------------------------------------------------------
// Depthwise 7x7 conv with reflect padding, one (b*c) plane tile per block.
// 38x40 f32 LDS tile, SGPR weights, 4 vertically-stacked outputs per thread.
// ---------------------------------------------------------------------------
#define TS  32
#define ITS 38

__global__ __launch_bounds__(256)
void dwconv7_kernel(const float* __restrict__ x,    // [256, 512, 512]
                    const float* __restrict__ km,   // [256, 49]
                    float* __restrict__ out)        // [256, 512, 512]
{
    __shared__ float tile[ITS][ITS + 2];            // bank-conflict pad

    const int bc  = blockIdx.z;
    const int ty0 = blockIdx.y * TS;
    const int tx0 = blockIdx.x * TS;
    const float* plane = x + (size_t)bc * (512 * 512);

    // per-plane 7x7 kernel: uniform index -> scalar loads
    float w[49];
#pragma unroll
    for (int i = 0; i < 49; ++i) w[i] = km[bc * 49 + i];

    for (int i = threadIdx.x; i < ITS * ITS; i += 256) {
        const int ly = i / ITS;
        const int lx = i - ly * ITS;
        int gy = ty0 + ly - 3;
        int gx = tx0 + lx - 3;
        gy = gy < 0 ? -gy : (gy > 511 ? 1022 - gy : gy);
        gx = gx < 0 ? -gx : (gx > 511 ? 1022 - gx : gx);
        tile[ly][lx] = plane[gy * 512 + gx];
    }
    __syncthreads();

    const int lx = threadIdx.x & 31;
    const int ly = (threadIdx.x >> 5) << 2;   // 4-row stack per thread
    float a0 = 0.f, a1 = 0.f, a2 = 0.f, a3 = 0.f;

#pragma unroll
    for (int kx = 0; kx < 7; ++kx) {
        float c[10];
#pragma unroll
        for (int r = 0; r < 10; ++r) c[r] = tile[ly + r][lx + kx];
#pragma unroll
        for (int ky = 0; ky < 7; ++ky) {
            const float wv = w[ky * 7 + kx];
            a0 = fmaf(c[ky],     wv, a0);
            a1 = fmaf(c[ky + 1], wv, a1);
            a2 = fmaf(c[ky + 2], wv, a2);
            a3 = fmaf(c[ky + 3], wv, a3);
        }
    }

    float* o = out + (size_t)bc * (512 * 512) + (size_t)(ty0 + ly) * 512 + (tx0 + lx);
    o[0]    = a0;
    o[512]  = a1;
    o[1024] = a2;
    o[1536] = a3;
}

// ---------------------------------------------------------------------------
extern "C" void kernel_launch(void* const* d_in, const int* in_sizes, int n_in,
                              void* d_out, int out_size, void* d_ws, size_t ws_size,
                              hipStream_t stream) {
    const float* x   = (const float*)d_in[0];   // [4,64,512,512]
    const float* rep = (const float*)d_in[1];   // [4,512,7,7]
    const float* wk  = (const float*)d_in[2];   // [64,512,7,7]
    const float* bk  = (const float*)d_in[3];   // [64]
    float* out = (float*)d_out;                 // [4,64,512,512]

    // workspace layout (floats), ~29 MB total; everything read is written first
    float* Apad    = (float*)d_ws;                          // 512*52*64
    float* Bpad    = Apad    + (size_t)512 * KPAD * 64;     // 512*52*208
    float* partial = Bpad    + (size_t)512 * KPAD * NPAD;   // 16*64*208
    float* keymap  = partial + (size_t)NSPLIT * 64 * NPAD;  // 4*64*49

    const int nA = 512 * KPAD * 64;
    const int nB = 512 * KPAD * NPAD;
    prep_a_kernel<<<dim3((nA + 255) / 256), dim3(256), 0, stream>>>(wk, Apad);
    prep_b_kernel<<<dim3((nB + 255) / 256), dim3(256), 0, stream>>>(rep, Bpad);

    // split-K WMMA GEMM: 52 C tiles x 16 K-splits, one wave each
    keymap_wmma_kernel<<<dim3(52 * NSPLIT), dim3(32), 0, stream>>>(Apad, Bpad, partial);

    reduce_kernel<<<dim3((64 * NPAD + 255) / 256), dim3(256), 0, stream>>>(partial, bk, keymap);

    // dominant stage: dynamic depthwise 7x7 conv over 256 planes of 512x512
    dwconv7_kernel<<<dim3(16, 16, 256), dim3(256), 0, stream>>>(x, keymap, out);
}